// MultiLevelDistiller_76278619177161
// MI455X (gfx1250) — compile-verified
//
#include <hip/hip_runtime.h>
#include <cstdint>

// Problem constants (from reference): h, old_h: [L,B,T,D] fp32, softmax over T.
#define L_ 4
#define B_ 32
#define T_ 2048
#define D_ 100
#define S_CHUNKS 4              // split of T across blocks
#define TC (T_ / S_CHUNKS)      // 512 rows per block
#define TS 32                   // rows staged per LDS tile
#define NTILES (TC / TS)        // 16 tiles per block
#define TSD (TS * D_)           // 3200 floats per tensor per tile (12.8 KB)
#define NF4 (TSD / 4)           // 800 float4 loads per tensor per tile
#define NTHREADS 256

// --- CDNA5 async global->LDS copy (ASYNCcnt-tracked) -----------------------
__device__ __forceinline__ void async_ld_f4(uint32_t lds_byte, uint32_t voff,
                                            const float* sbase) {
  // GVS mode: mem_addr = SADDR(64, sgpr pair) + VADDR(32, per-lane); dest = LDS byte addr
  asm volatile("global_load_async_to_lds_b128 %0, %1, %2"
               :: "v"(lds_byte), "v"(voff), "s"(sbase) : "memory");
}
#define WAIT_ASYNC(N) asm volatile("s_wait_asynccnt " #N ::: "memory")

// Kernel 1: per-(l,b,chunk) block streams [TC x D] of h and old_h through a
// double-buffered LDS pipeline; two-pass softmax per tile; accumulates
// online-mergeable 6-float states per class.
__global__ __launch_bounds__(NTHREADS)
void kl_partial(const float* __restrict__ h, const float* __restrict__ oh,
                float* __restrict__ part) {
  __shared__ float smem[4 * TSD];     // 51.2 KB: [b0_h | b0_p | b1_h | b1_p]

  const int x   = threadIdx.x;        // 0..31, lanes < 25 own 4 classes each
  const int y   = threadIdx.y;        // 0..7  (one wave per y)
  const int tid = y * 32 + x;
  const int s   = blockIdx.x;         // T-chunk
  const int lb  = blockIdx.y;         // l*B + b

  const size_t base = ((size_t)lb * T_ + (size_t)s * TC) * D_;
  const float* gh = h  + base;
  const float* gp = oh + base;

  const uint32_t l0 = (uint32_t)(uintptr_t)smem;  // LDS byte offset (low 32 bits)

  // State per class k: m_p, Zp, A=sum e*xp, B=sum e*xq, m_q, Zq
  float mp[4], zp[4], aa[4], bb[4], mq[4], zq[4];
#pragma unroll
  for (int k = 0; k < 4; ++k) {
    mp[k] = -__builtin_inff(); mq[k] = -__builtin_inff();
    zp[k] = 0.f; aa[k] = 0.f; bb[k] = 0.f; zq[k] = 0.f;
  }
  const int  d0     = 4 * x;
  const bool active = (x < 25);       // 25 lanes * 4 classes = 100

  // Issue one tile's async copies into buffer `buf`.
  // Per-wave instruction count: wave y==0 -> 8, waves y=1..7 -> 6 (uniform).
  auto issue = [&](int tile, int buf) {
    const float*   th = gh + (size_t)tile * TSD;
    const float*   tp = gp + (size_t)tile * TSD;
    const uint32_t lh = l0 + (uint32_t)buf * (2u * TSD * 4u);
    const uint32_t lp = lh + TSD * 4u;
    for (int i = tid; i < NF4; i += NTHREADS) {   // contiguous 16B per lane
      const uint32_t off = (uint32_t)i * 16u;
      async_ld_f4(lh + off, off, th);
      async_ld_f4(lp + off, off, tp);
    }
  };

  issue(0, 0);
  for (int tile = 0; tile < NTILES; ++tile) {
    const int buf = tile & 1;
    if (tile + 1 < NTILES) {
      issue(tile + 1, buf ^ 1);       // overlap next DMA with this compute
      if (y == 0) { WAIT_ASYNC(8); } else { WAIT_ASYNC(6); }  // current tile done
    } else {
      WAIT_ASYNC(0);
    }
    __syncthreads();                  // all waves' loads for `buf` complete

    if (active) {
      const float* s_h = &smem[buf * 2 * TSD];
      const float* s_p = s_h + TSD;
      // Pass 1: tile max per class over this thread's rows
      float tmq[4], tmx[4];
#pragma unroll
      for (int k = 0; k < 4; ++k) { tmq[k] = -__builtin_inff(); tmx[k] = -__builtin_inff(); }
      for (int r = y; r < TS; r += 8) {
        const float* rh = &s_h[r * D_ + d0];
        const float* rp = &s_p[r * D_ + d0];
#pragma unroll
        for (int k = 0; k < 4; ++k) {
          tmq[k] = fmaxf(tmq[k], rh[k]);
          tmx[k] = fmaxf(tmx[k], rp[k]);
        }
      }
      // Rescale running state once per tile (not per element)
#pragma unroll
      for (int k = 0; k < 4; ++k) {
        const float nmp = fmaxf(mp[k], tmx[k]);
        const float sc  = __expf(mp[k] - nmp);
        zp[k] *= sc; aa[k] *= sc; bb[k] *= sc; mp[k] = nmp;
        const float nmq = fmaxf(mq[k], tmq[k]);
        zq[k] *= __expf(mq[k] - nmq); mq[k] = nmq;
      }
      // Pass 2: exp-sums (2 transcendentals per element pair)
      for (int r = y; r < TS; r += 8) {
        const float* rh = &s_h[r * D_ + d0];
        const float* rp = &s_p[r * D_ + d0];
#pragma unroll
        for (int k = 0; k < 4; ++k) {
          const float xq = rh[k], xp = rp[k];
          const float e  = __expf(xp - mp[k]);
          zp[k] += e; aa[k] += e * xp; bb[k] += e * xq;
          zq[k] += __expf(xq - mq[k]);
        }
      }
    }
    __syncthreads();                  // protect buffers before reuse
  }

  // Merge the 8 y-partitions per class via LDS (6144 floats; buffer 0 space,
  // last read two tiles ago, so safe after the final barrier above).
  if (active) {
    float* o = &smem[tid * 24];
#pragma unroll
    for (int k = 0; k < 4; ++k) {
      o[k] = mp[k]; o[4 + k] = zp[k]; o[8 + k] = aa[k];
      o[12 + k] = bb[k]; o[16 + k] = mq[k]; o[20 + k] = zq[k];
    }
  }
  __syncthreads();
  if (y == 0 && active) {
    for (int yy = 1; yy < 8; ++yy) {
      const float* o = &smem[(yy * 32 + x) * 24];
#pragma unroll
      for (int k = 0; k < 4; ++k) {
        const float omp = o[k], ozp = o[4 + k], oaa = o[8 + k],
                    obb = o[12 + k], omq = o[16 + k], ozq = o[20 + k];
        const float nm = fmaxf(mp[k], omp);
        const float s1 = __expf(mp[k] - nm), s2 = __expf(omp - nm);
        zp[k] = zp[k] * s1 + ozp * s2;
        aa[k] = aa[k] * s1 + oaa * s2;
        bb[k] = bb[k] * s1 + obb * s2;
        mp[k] = nm;
        const float nmq = fmaxf(mq[k], omq);
        zq[k] = zq[k] * __expf(mq[k] - nmq) + ozq * __expf(omq - nmq);
        mq[k] = nmq;
      }
    }
    const int blk = lb * S_CHUNKS + s;
#pragma unroll
    for (int k = 0; k < 4; ++k) {
      float* o = part + ((size_t)blk * D_ + (d0 + k)) * 6;
      o[0] = mp[k]; o[1] = zp[k]; o[2] = aa[k];
      o[3] = bb[k]; o[4] = mq[k]; o[5] = zq[k];
    }
  }
}

// Kernel 2: merge chunk states per column, KL per column, batchmean over B,
// apply softmax(layer_weights) x softmax(class_weights), reduce to scalar.
__global__ __launch_bounds__(512)
void kl_finalize(const float* __restrict__ part, const float* __restrict__ lwv,
                 const float* __restrict__ cwv, float* __restrict__ out) {
  __shared__ float red[512];
  const int tid = threadIdx.x;
  float contrib = 0.f;
  if (tid < L_ * D_) {
    const int l = tid / D_, d = tid % D_;
    float acc = 0.f;
    for (int b = 0; b < B_; ++b) {
      const int lb = l * B_ + b;
      const float* p = part + ((size_t)(lb * S_CHUNKS) * D_ + d) * 6;
      float mp = p[0], zp = p[1], aa = p[2], bb = p[3], mq = p[4], zq = p[5];
      for (int sc = 1; sc < S_CHUNKS; ++sc) {
        const float* q = part + ((size_t)(lb * S_CHUNKS + sc) * D_ + d) * 6;
        const float omp = q[0], ozp = q[1], oaa = q[2],
                    obb = q[3], omq = q[4], ozq = q[5];
        const float nm = fmaxf(mp, omp);
        const float s1 = __expf(mp - nm), s2 = __expf(omp - nm);
        zp = zp * s1 + ozp * s2;
        aa = aa * s1 + oaa * s2;
        bb = bb * s1 + obb * s2;
        mp = nm;
        const float nmq = fmaxf(mq, omq);
        zq = zq * __expf(mq - nmq) + ozq * __expf(omq - nmq);
        mq = nmq;
      }
      // KL(p||q) for this column
      acc += (aa - bb) / zp - mp - __logf(zp) + mq + __logf(zq);
    }
    acc *= (1.0f / B_);               // batchmean

    float lmax = -__builtin_inff();
    for (int i = 0; i < L_; ++i) lmax = fmaxf(lmax, lwv[i]);
    float lsum = 0.f;
    for (int i = 0; i < L_; ++i) lsum += __expf(lwv[i] - lmax);
    const float lw = __expf(lwv[l] - lmax) / lsum;

    float cmax = -__builtin_inff();
    for (int i = 0; i < D_; ++i) cmax = fmaxf(cmax, cwv[i]);
    float csum = 0.f;
    for (int i = 0; i < D_; ++i) csum += __expf(cwv[i] - cmax);
    const float cw = __expf(cwv[d] - cmax) / csum;

    contrib = lw * cw * acc;
  }
  red[tid] = contrib;
  __syncthreads();
#pragma unroll
  for (int off = 256; off > 0; off >>= 1) {
    if (tid < off) red[tid] += red[tid + off];
    __syncthreads();
  }
  if (tid == 0) out[0] = red[0];
}

extern "C" void kernel_launch(void* const* d_in, const int* in_sizes, int n_in,
                              void* d_out, int out_size, void* d_ws, size_t ws_size,
                              hipStream_t stream) {
  const float* h  = (const float*)d_in[0];
  const float* oh = (const float*)d_in[1];
  const float* lw = (const float*)d_in[2];
  const float* cw = (const float*)d_in[3];
  float* part = (float*)d_ws;  // needs L*B*S_CHUNKS*D*6*4 = 1,228,800 bytes

  kl_partial<<<dim3(S_CHUNKS, L_ * B_), dim3(32, 8), 0, stream>>>(h, oh, part);
  kl_finalize<<<1, 512, 0, stream>>>(part, lw, cw, (float*)d_out);
}